// DroneGNNPolicy_71983651881176
// MI455X (gfx1250) — compile-verified
//
#include <hip/hip_runtime.h>
#include <hip/hip_bf16.h>

typedef __attribute__((ext_vector_type(2))) float v2f;
typedef __attribute__((ext_vector_type(8))) float v8f;

#define LRELU_SLOPE 0.2f

// ---------- helpers ----------

__device__ __forceinline__ void atomicMaxF(float* addr, float v) {
    // works for mixed-sign values when addr initialized to -inf:
    // positive floats ordered as signed ints; negative floats reverse-ordered as uints
    if (v >= 0.f) atomicMax((int*)addr, __float_as_int(v));
    else          atomicMin((unsigned int*)addr, __float_as_uint(v));
}

__device__ __forceinline__ void atomAddF(float* p, float v) {
    __hip_atomic_fetch_add(p, v, __ATOMIC_RELAXED, __HIP_MEMORY_SCOPE_AGENT);
}

__device__ __forceinline__ float leaky(float a) {
    return a > 0.f ? a : LRELU_SLOPE * a;
}

// ---------- fill ----------

__global__ void fill_k(float* __restrict__ p, float v, long long n) {
    long long t = (long long)blockIdx.x * blockDim.x + threadIdx.x;
    if (t < n) p[t] = v;
}

// ---------- layer-1 node transform: h1 = x@W1 ; per-head attention dots ----------
// one block (256 threads) per node; F_in = 6, H*C = 256

__global__ void transform1_k(const float* __restrict__ x, const float* __restrict__ W1,
                             const float* __restrict__ att_s, const float* __restrict__ att_d,
                             float* __restrict__ h1, float* __restrict__ as1,
                             float* __restrict__ ad1, int N) {
    __shared__ float xs[6];
    __shared__ float ps[256];
    __shared__ float pd[256];
    int n = blockIdx.x;
    int c = threadIdx.x;                      // output channel 0..255
    if (c < 6) xs[c] = x[(size_t)n * 6 + c];
    __syncthreads();
    float h = 0.f;
#pragma unroll
    for (int k = 0; k < 6; ++k) h += xs[k] * W1[k * 256 + c];
    h1[(size_t)n * 256 + c] = h;
    ps[c] = h * att_s[c];                     // att_src1 flat [4*64]
    pd[c] = h * att_d[c];
    __syncthreads();
    int cc = c & 63;
    for (int off = 32; off > 0; off >>= 1) {
        if (cc < off) { ps[c] += ps[c + off]; pd[c] += pd[c + off]; }
        __syncthreads();
    }
    if (cc == 0) {
        int head = c >> 6;
        as1[(size_t)n * 4 + head] = ps[c];
        ad1[(size_t)n * 4 + head] = pd[c];
    }
}

// ---------- layer-2 per-node attention dots (H=1, C=64) ----------

__global__ void node_att2_k(const float* __restrict__ h2, const float* __restrict__ att_s,
                            const float* __restrict__ att_d, float* __restrict__ as2,
                            float* __restrict__ ad2, int N) {
    __shared__ float ps[64];
    __shared__ float pd[64];
    int n = blockIdx.x;
    int c = threadIdx.x;                      // 0..63
    float h = h2[(size_t)n * 64 + c];
    ps[c] = h * att_s[c];
    pd[c] = h * att_d[c];
    __syncthreads();
    for (int off = 32; off > 0; off >>= 1) {
        if (c < off) { ps[c] += ps[c + off]; pd[c] += pd[c + off]; }
        __syncthreads();
    }
    if (c == 0) { as2[n] = ps[0]; ad2[n] = pd[0]; }
}

// ---------- edge kernels (self-loops appended virtually: e in [E, E+N) => s=d=e-E) ----------

template <int H>
__global__ void edge_max_k(const long long* __restrict__ ei, int E, int N,
                           const float* __restrict__ as, const float* __restrict__ ad,
                           float* __restrict__ m) {
    long long t = (long long)blockIdx.x * blockDim.x + threadIdx.x;
    long long total = (long long)(E + N) * H;
    if (t >= total) return;
    int hh = (int)(t % H);
    long long e = t / H;
    int s, d;
    if (e < E) { s = (int)ei[e]; d = (int)ei[(long long)E + e]; }
    else       { s = d = (int)(e - E); }
    float a = leaky(as[(size_t)s * H + hh] + ad[(size_t)d * H + hh]);
    atomicMaxF(&m[(size_t)d * H + hh], a);
}

template <int H>
__global__ void edge_sum_k(const long long* __restrict__ ei, int E, int N,
                           const float* __restrict__ as, const float* __restrict__ ad,
                           const float* __restrict__ m, float* __restrict__ dn) {
    long long t = (long long)blockIdx.x * blockDim.x + threadIdx.x;
    long long total = (long long)(E + N) * H;
    if (t >= total) return;
    int hh = (int)(t % H);
    long long e = t / H;
    int s, d;
    if (e < E) { s = (int)ei[e]; d = (int)ei[(long long)E + e]; }
    else       { s = d = (int)(e - E); }
    float a = leaky(as[(size_t)s * H + hh] + ad[(size_t)d * H + hh]);
    atomAddF(&dn[(size_t)d * H + hh], expf(a - m[(size_t)d * H + hh]));
}

template <int H, int C>
__global__ void edge_msg_k(const long long* __restrict__ ei, int E, int N,
                           const float* __restrict__ as, const float* __restrict__ ad,
                           const float* __restrict__ m, const float* __restrict__ dn,
                           const float* __restrict__ hf, float* __restrict__ out) {
    long long t = (long long)blockIdx.x * blockDim.x + threadIdx.x;
    long long total = (long long)(E + N) * H;
    if (t >= total) return;
    int hh = (int)(t % H);
    long long e = t / H;
    int s, d;
    if (e < E) { s = (int)ei[e]; d = (int)ei[(long long)E + e]; }
    else       { s = d = (int)(e - E); }
    float a = leaky(as[(size_t)s * H + hh] + ad[(size_t)d * H + hh]);
    float w = expf(a - m[(size_t)d * H + hh]) / fmaxf(dn[(size_t)d * H + hh], 1e-16f);
    const float4* hs = (const float4*)(hf + (size_t)s * (H * C) + (size_t)hh * C);
    float* od = out + (size_t)d * (H * C) + (size_t)hh * C;
#pragma unroll 4
    for (int i = 0; i < C / 4; ++i) {
        float4 v = hs[i];
        atomAddF(od + 4 * i + 0, v.x * w);
        atomAddF(od + 4 * i + 1, v.y * w);
        atomAddF(od + 4 * i + 2, v.z * w);
        atomAddF(od + 4 * i + 3, v.w * w);
    }
}

// ---------- bias + ELU (in place) ----------

template <int HC>
__global__ void bias_elu_k(float* __restrict__ o, const float* __restrict__ b, long long total) {
    long long t = (long long)blockIdx.x * blockDim.x + threadIdx.x;
    if (t >= total) return;
    float v = o[t] + b[(int)(t % HC)];
    o[t] = v > 0.f ? v : (expf(v) - 1.f);
}

// ---------- f32 WMMA GEMM: D[M,64] = A[M,K] @ B[K,64] ----------
// Block = 256 threads = 8 waves. Wave w owns one 16-row strip of D across all 64
// columns: 4 accumulators -> one A-fragment load feeds 4 v_wmma_f32_16x16x4_f32.
//
// B is staged in a fragment-ready pair-interleaved LDS layout: pair-row kp holds
// float2 {B[2kp][c], B[2kp+1][c]} for c=0..63, so each B fragment is ONE 8-byte
// ds_load_b64 directly into an even-aligned VGPR pair (no register shuffles).
// Pair-row stride = 160 floats (128 used + 32 pad): the two half-waves read
// adjacent pair-rows offset by 160 % 64 == 32 banks -> disjoint 32-bank sets,
// conflict-free b64 loads.
//
// V_WMMA_F32_16X16X4_F32 fragment layouts (ISA 7.12.2):
//   A 16x4: lane<16 -> row=lane, {K=k0,k0+1}; lane>=16 -> row=lane-16, {K=k0+2,k0+3}
//   B 4x16: lane<16 -> col=lane, {K=k0,k0+1}; lane>=16 -> col=lane-16, {K=k0+2,k0+3}
//   C/D  : vgpr i -> row = i + 8*(lane>=16), col = lane&15
// A-row clamp is safe: A row r only contributes to D row r, and rows >= M are
// never stored.

template <int K, bool RELU_BIAS>
__global__ __launch_bounds__(256) void wmma_gemm_f32_k(
    const float* __restrict__ A, const float* __restrict__ B,
    const float* __restrict__ bias, float* __restrict__ D, int M) {
    constexpr int NC = 64;
    constexpr int LDBP = 160;                     // floats per pair-row (128 + 32 pad)
    __shared__ float Bs[(K / 2) * LDBP];
    for (int idx = threadIdx.x; idx < (K / 2) * NC; idx += 256) {
        int kp = idx >> 6, cc = idx & 63;
        v2f p;
        p.x = B[kp * 128 + cc];                   // B[2kp][cc]
        p.y = B[kp * 128 + 64 + cc];              // B[2kp+1][cc]
        *(v2f*)&Bs[kp * LDBP + 2 * cc] = p;
    }
    __syncthreads();

    int wave = (blockIdx.x << 3) + (threadIdx.x >> 5);
    int lane = threadIdx.x & 31;
    if (wave * 16 >= M) return;                   // uniform per wave
    int half = lane >> 4;
    int l = lane & 15;
    int arow = wave * 16 + l;
    int arow_c = arow < M ? arow : (M - 1);       // clamp (see note above)
    const float* Arow = A + (size_t)arow_c * K;

    v8f acc0 = {}, acc1 = {}, acc2 = {}, acc3 = {};
    for (int k0 = 0; k0 < K; k0 += 4) {
        v2f a = *(const v2f*)(Arow + k0 + 2 * half);     // 8B-aligned
        const v2f* brow = (const v2f*)&Bs[((k0 >> 1) + half) * LDBP];
        v2f b0 = brow[l];
        v2f b1 = brow[16 + l];
        v2f b2 = brow[32 + l];
        v2f b3 = brow[48 + l];
        acc0 = __builtin_amdgcn_wmma_f32_16x16x4_f32(false, a, false, b0, (short)0, acc0, false, false);
        acc1 = __builtin_amdgcn_wmma_f32_16x16x4_f32(false, a, false, b1, (short)0, acc1, false, false);
        acc2 = __builtin_amdgcn_wmma_f32_16x16x4_f32(false, a, false, b2, (short)0, acc2, false, false);
        acc3 = __builtin_amdgcn_wmma_f32_16x16x4_f32(false, a, false, b3, (short)0, acc3, false, false);
    }

    int rbase = wave * 16 + 8 * half;
    bool full = (wave * 16 + 16) <= M;            // uniform per wave
    if (full) {
#pragma unroll
        for (int i = 0; i < 8; ++i) {
            float* drow = D + (size_t)(rbase + i) * NC;
            float v0 = acc0[i], v1 = acc1[i], v2 = acc2[i], v3 = acc3[i];
            if constexpr (RELU_BIAS) {
                v0 = fmaxf(v0 + bias[l], 0.f);
                v1 = fmaxf(v1 + bias[16 + l], 0.f);
                v2 = fmaxf(v2 + bias[32 + l], 0.f);
                v3 = fmaxf(v3 + bias[48 + l], 0.f);
            }
            drow[l] = v0; drow[16 + l] = v1; drow[32 + l] = v2; drow[48 + l] = v3;
        }
    } else {
#pragma unroll
        for (int i = 0; i < 8; ++i) {
            int row = rbase + i;
            if (row < M) {
                float* drow = D + (size_t)row * NC;
                float v0 = acc0[i], v1 = acc1[i], v2 = acc2[i], v3 = acc3[i];
                if constexpr (RELU_BIAS) {
                    v0 = fmaxf(v0 + bias[l], 0.f);
                    v1 = fmaxf(v1 + bias[16 + l], 0.f);
                    v2 = fmaxf(v2 + bias[32 + l], 0.f);
                    v3 = fmaxf(v3 + bias[48 + l], 0.f);
                }
                drow[l] = v0; drow[16 + l] = v1; drow[32 + l] = v2; drow[48 + l] = v3;
            }
        }
    }
}

// ---------- fc2 + tanh head ----------

__global__ void fc2_tanh_k(const float* __restrict__ z1, const float* __restrict__ w,
                           const float* __restrict__ b, float* __restrict__ out, int nd) {
    int t = blockIdx.x * blockDim.x + threadIdx.x;
    if (t >= nd * 2) return;
    int r = t >> 1, o = t & 1;
    const float* zr = z1 + (size_t)r * 64;
    float s = b[o];
#pragma unroll
    for (int k = 0; k < 64; ++k) s += zr[k] * w[k * 2 + o];
    out[t] = tanhf(s) * 2.0f;
}

// ---------- launch ----------

static inline unsigned gdiv(long long n, int b) { return (unsigned)((n + b - 1) / b); }

extern "C" void kernel_launch(void* const* d_in, const int* in_sizes, int n_in,
                              void* d_out, int out_size, void* d_ws, size_t ws_size,
                              hipStream_t stream) {
    const float*     x      = (const float*)d_in[0];
    const long long* ei     = (const long long*)d_in[1];
    const float*     W1     = (const float*)d_in[3];
    const float*     att_s1 = (const float*)d_in[4];
    const float*     att_d1 = (const float*)d_in[5];
    const float*     b1     = (const float*)d_in[6];
    const float*     W2     = (const float*)d_in[7];
    const float*     att_s2 = (const float*)d_in[8];
    const float*     att_d2 = (const float*)d_in[9];
    const float*     b2     = (const float*)d_in[10];
    const float*     fc1w   = (const float*)d_in[11];
    const float*     fc1b   = (const float*)d_in[12];
    const float*     fc2w   = (const float*)d_in[13];
    const float*     fc2b   = (const float*)d_in[14];
    float*           out    = (float*)d_out;

    const int N  = in_sizes[0] / 6;
    const int E  = in_sizes[1] / 2;
    const int nd = out_size / 2;              // drones are rows [0, nd)
    const size_t Nz = (size_t)N;

    float* ws   = (float*)d_ws;
    float* h1   = ws;                         // [N,256]
    float* out1 = ws + Nz * 256;              // [N,256]
    float* as1  = ws + Nz * 512;              // [N,4]
    float* ad1  = as1 + Nz * 4;
    float* m1   = ad1 + Nz * 4;
    float* dn1  = m1 + Nz * 4;
    float* as2  = dn1 + Nz * 4;               // [N]
    float* ad2  = as2 + Nz;
    float* m2   = ad2 + Nz;
    float* dn2  = m2 + Nz;
    // aliases into the h1 region (h1 is dead after layer-1 messaging):
    float* h2   = ws;                         // [N,64]
    float* out2 = ws + Nz * 64;               // [N,64]
    float* z1   = ws + Nz * 128;              // [nd,64]

    const int B = 256;
    const long long EN1 = (long long)(E + N) * 4;
    const long long EN2 = (long long)(E + N);

    // ---- layer 1 ----
    fill_k<<<gdiv(Nz * 256, B), B, 0, stream>>>(out1, 0.f, (long long)Nz * 256);
    fill_k<<<gdiv(Nz * 4, B), B, 0, stream>>>(m1, -INFINITY, (long long)Nz * 4);
    fill_k<<<gdiv(Nz * 4, B), B, 0, stream>>>(dn1, 0.f, (long long)Nz * 4);

    transform1_k<<<N, 256, 0, stream>>>(x, W1, att_s1, att_d1, h1, as1, ad1, N);
    edge_max_k<4><<<gdiv(EN1, B), B, 0, stream>>>(ei, E, N, as1, ad1, m1);
    edge_sum_k<4><<<gdiv(EN1, B), B, 0, stream>>>(ei, E, N, as1, ad1, m1, dn1);
    edge_msg_k<4, 64><<<gdiv(EN1, B), B, 0, stream>>>(ei, E, N, as1, ad1, m1, dn1, h1, out1);
    bias_elu_k<256><<<gdiv(Nz * 256, B), B, 0, stream>>>(out1, b1, (long long)Nz * 256);

    // ---- h2 = out1 @ W2  (WMMA f32, K=256) ----
    {
        int waves = (N + 15) / 16;
        wmma_gemm_f32_k<256, false><<<gdiv(waves, 8), B, 0, stream>>>(out1, W2, nullptr, h2, N);
    }

    // ---- layer 2 ----
    fill_k<<<gdiv(Nz * 64, B), B, 0, stream>>>(out2, 0.f, (long long)Nz * 64);
    fill_k<<<gdiv(Nz, B), B, 0, stream>>>(m2, -INFINITY, (long long)Nz);
    fill_k<<<gdiv(Nz, B), B, 0, stream>>>(dn2, 0.f, (long long)Nz);

    node_att2_k<<<N, 64, 0, stream>>>(h2, att_s2, att_d2, as2, ad2, N);
    edge_max_k<1><<<gdiv(EN2, B), B, 0, stream>>>(ei, E, N, as2, ad2, m2);
    edge_sum_k<1><<<gdiv(EN2, B), B, 0, stream>>>(ei, E, N, as2, ad2, m2, dn2);
    edge_msg_k<1, 64><<<gdiv(EN2, B), B, 0, stream>>>(ei, E, N, as2, ad2, m2, dn2, h2, out2);
    bias_elu_k<64><<<gdiv(Nz * 64, B), B, 0, stream>>>(out2, b2, (long long)Nz * 64);

    // ---- MLP head: z1 = relu(out2[0:nd] @ fc1_w + fc1_b)  (WMMA f32, K=64) ----
    {
        int waves = (nd + 15) / 16;
        wmma_gemm_f32_k<64, true><<<gdiv(waves, 8), B, 0, stream>>>(out2, fc1w, fc1b, z1, nd);
    }
    fc2_tanh_k<<<gdiv((long long)nd * 2, B), B, 0, stream>>>(z1, fc2w, fc2b, out, nd);
}